// Type4_21380347200016
// MI455X (gfx1250) — compile-verified
//
#include <hip/hip_runtime.h>

typedef __attribute__((ext_vector_type(2))) float v2f;
typedef __attribute__((ext_vector_type(8))) float v8f;

#define B_DIM 32
#define C_DIM 12
#define ROWS  (B_DIM * C_DIM)   // 384 (b,c) rows
#define TILES (ROWS / 16)       // 24 WMMA tiles -> 24 waves

// One workgroup does everything:
//   stage 1: 24 waves, each does one V_WMMA_F32_16X16X4_F32 tile of the
//            384x3 = (384x4pad) @ (4pad x 16pad) GEMM, spills 3 cols to LDS
//   stage 2: 384 threads do the 3-wide softmax per row in LDS
//   stage 3: 96 threads reduce over c, apply sigmoid, scale by 12
__global__ __launch_bounds__(768) void fused_softmax_gemm_kernel(
    const float* __restrict__ x,     // (32,12,3,73) f32, only [:, :, 0, :3] used
    const float* __restrict__ thr,   // (12,3,3) f32
    const float* __restrict__ J,     // (3,3)    f32
    float* __restrict__ out)         // (32,3)   f32
{
    __shared__ float sres[ROWS * 4]; // [row][k] padded to 4 for banking

    const int tid  = threadIdx.x;
    const int lane = tid & 31;
    const int wave = tid >> 5;        // tile index 0..23
    const int half = lane >> 4;       // 0: K={0,1}; 1: K={2,3}
    const int l15  = lane & 15;

    // ---------------- Build A operand (16x4 f32, rows = 16 (b,c) pairs) ---
    // A[m,k]: lanes 0-15 hold K=0 (x) and K=1 (y) for M=lane;
    //         lanes 16-31 hold K=2 (x) and K=3 (y) for M=lane-16.
    const int row = wave * 16 + l15;
    const int b   = row / C_DIM;
    const int c   = row % C_DIM;
    const float* xp = x   + b * (C_DIM * 3 * 73) + c * (3 * 73); // x[b,c,0,:]
    const float* tp = thr + c * 9;                               // thr[c,:,:]

    v2f a;
    if (half == 0) {
        a.x = 3.0f * xp[0] - (tp[0] + tp[3] + tp[6]);   // j = 0
        a.y = 3.0f * xp[1] - (tp[1] + tp[4] + tp[7]);   // j = 1
    } else {
        a.x = 3.0f * xp[2] - (tp[2] + tp[5] + tp[8]);   // j = 2
        a.y = 0.0f;                                     // j = 3 (pad)
    }

    // ---------------- Build B operand (4x16 f32): B[k_row, n] -------------
    // lanes 0-15: N=lane, K=0 (x) / K=1 (y); lanes 16-31: K=2 (x) / K=3 (y)
    v2f bm;
    bm.x = 0.0f;
    bm.y = 0.0f;
    if (l15 < 3) {
        if (half == 0) {
            bm.x = J[0 * 3 + l15];   // K = 0
            bm.y = J[1 * 3 + l15];   // K = 1
        } else {
            bm.x = J[2 * 3 + l15];   // K = 2
            bm.y = 0.0f;             // K = 3 (pad)
        }
    }

    // ---------------- WMMA: D = A x B + 0 ---------------------------------
    // (neg_a, A, neg_b, B, c_mod, C, reuse_a, reuse_b)
    v8f d = {};
    d = __builtin_amdgcn_wmma_f32_16x16x4_f32(
        false, a, false, bm, (short)0, d, false, false);

    // ---------------- Spill useful columns (n < 3) of D to LDS ------------
    // D layout: lanes 0-15 -> N=lane, M=vgpr; lanes 16-31 -> N=lane-16, M=vgpr+8
    if (l15 < 3) {
        const int mbase = wave * 16 + half * 8;
        #pragma unroll
        for (int v = 0; v < 8; ++v)
            sres[(mbase + v) * 4 + l15] = d[v];
    }
    __syncthreads();

    // ---------------- Per-row softmax over k = 0..2 ------------------------
    if (tid < ROWS) {
        float r0 = sres[tid * 4 + 0];
        float r1 = sres[tid * 4 + 1];
        float r2 = sres[tid * 4 + 2];
        float m  = fmaxf(r0, fmaxf(r1, r2));
        float e0 = __expf(r0 - m);
        float e1 = __expf(r1 - m);
        float e2 = __expf(r2 - m);
        // single v_rcp_f32 instead of the IEEE div_scale/div_fmas sequence
        float inv = __builtin_amdgcn_rcpf(e0 + e1 + e2);
        sres[tid * 4 + 0] = e0 * inv;
        sres[tid * 4 + 1] = e1 * inv;
        sres[tid * 4 + 2] = e2 * inv;
    }
    __syncthreads();

    // ---------------- Sum over c, sigmoid, scale by C_DIM ------------------
    if (tid < B_DIM * 3) {
        const int ob = tid / 3;
        const int k  = tid % 3;
        float s = 0.0f;
        #pragma unroll
        for (int cc = 0; cc < C_DIM; ++cc)
            s += sres[(ob * C_DIM + cc) * 4 + k];
        // sigmoid via v_rcp_f32: 12 * 1/(1+exp(-s))
        out[ob * 3 + k] =
            (float)C_DIM * __builtin_amdgcn_rcpf(1.0f + __expf(-s));
    }
}

extern "C" void kernel_launch(void* const* d_in, const int* in_sizes, int n_in,
                              void* d_out, int out_size, void* d_ws, size_t ws_size,
                              hipStream_t stream) {
    (void)in_sizes; (void)n_in; (void)out_size; (void)d_ws; (void)ws_size;
    const float* x   = (const float*)d_in[0];
    // d_in[1] = attention: dead in the reference, ignored
    const float* thr = (const float*)d_in[2];
    const float* J   = (const float*)d_in[3];
    float* out = (float*)d_out;

    fused_softmax_gemm_kernel<<<1, 768, 0, stream>>>(x, thr, J, out);
}